// AcidSynth_77816217468987
// MI455X (gfx1250) — compile-verified
//
#include <hip/hip_runtime.h>
#include <hip/hip_bf16.h>

typedef __attribute__((ext_vector_type(2))) float v2f;
typedef __attribute__((ext_vector_type(8))) float v8f;
typedef unsigned int u32;
typedef __attribute__((ext_vector_type(4))) u32 v4u;
typedef __attribute__((ext_vector_type(4))) int v4i;
typedef __attribute__((ext_vector_type(8))) int v8i;

constexpr int   NSAMP   = 131072;
constexpr int   BATCH   = 32;
constexpr int   CHUNK   = 128;             // samples per chunk (serial chain)
constexpr int   NCH     = NSAMP / CHUNK;   // 1024 chunks per row
constexpr int   NCHUNKS = BATCH * NCH;     // 32768
constexpr int   RPB     = 128;             // chunks (rows) per block
constexpr int   TILE    = 32;              // samples staged per chunk per TDM op
constexpr int   NTILES  = CHUNK / TILE;    // 4
constexpr int   PITCH   = TILE + 1;        // 33: coprime with 64 banks
constexpr float SRf     = 48000.0f;
constexpr float TWOPI   = 6.28318530717958647692f;
constexpr float MINW    = 0.00261799387799149436f;   // 2*pi*20/48000
constexpr float LN_WR   = 5.99146454710798198f;      // ln(8000/20)
constexpr float MINQ    = 0.7071f;
constexpr float LN_QR   = 2.42602468f;               // ln(8.0/0.7071)

#if __has_builtin(__builtin_amdgcn_tensor_load_to_lds) && \
    __has_builtin(__builtin_amdgcn_tensor_store_from_lds) && \
    __has_builtin(__builtin_amdgcn_s_wait_tensorcnt)
#define HAVE_TDM 1
#else
#define HAVE_TDM 0
#endif

__device__ __forceinline__ u32 ldsoff(const void* p) {
    // flat shared address: LDS byte offset lives in the low 32 bits
    return (u32)(unsigned long long)p;
}

#if HAVE_TDM
// ---------------------------------------------------------------------------
// TDM iterate-mode descriptor: per iteration copy TILE contiguous f32 between
// global and LDS, then global_addr += CHUNK elems, lds_addr += PITCH elems.
// RPB iterations total. Exactly a strided-2D <-> pitched-LDS tile copy.
// ---------------------------------------------------------------------------
__device__ __forceinline__ void tdm_issue(bool store, u32 lds_byte, const void* gaddr) {
    unsigned long long ga = (unsigned long long)gaddr;
    v4u g0 = { 1u,                                    // count=1 (valid D#)
               lds_byte,                              // lds_addr (bytes)
               (u32)(ga & 0xFFFFFFFFull),             // global_addr[31:0]
               (u32)((ga >> 32) & 0x01FFFFFFull) | (2u << 30) };  // type=2
    v8i g1 = { (2 << 16) | (1 << 19),                 // data_size=4B, iterate_enable
               0,                                     // no atomic barrier; dim0 lo16=0
               16,                                    // tensor_dim0=1<<20 hi16; dim1 lo16=0
               16 | (TILE << 16),                     // tensor_dim1 hi16; tile_dim0=TILE
               1,                                     // tile_dim1=1, tile_dim2=0
               CHUNK,                                 // tensor_dim0_stride lo32
               0, 0 };
    v4i g2 = { 0,                                     // tensor_dim2
               PITCH,                                 // lds_addr_increment (elems)
               CHUNK,                                 // global_addr_increment lo32 (elems)
               (RPB - 1) << 16 };                     // iterate_count: RPB iterations
    v4i g3 = { 0, 0, 0, 0 };
#if defined(__clang_major__) && (__clang_major__ >= 23)
    v8i g4 = { 0, 0, 0, 0, 0, 0, 0, 0 };
    if (store) __builtin_amdgcn_tensor_store_from_lds(g0, g1, g2, g3, g4, 0);
    else       __builtin_amdgcn_tensor_load_to_lds  (g0, g1, g2, g3, g4, 0);
#else
    if (store) __builtin_amdgcn_tensor_store_from_lds(g0, g1, g2, g3, 0);
    else       __builtin_amdgcn_tensor_load_to_lds  (g0, g1, g2, g3, 0);
#endif
}

__device__ __forceinline__ void wait_tensor(int n) {
    switch (n) {                      // immediate operand must be constant
        case 0: __builtin_amdgcn_s_wait_tensorcnt(0); break;
        case 1: __builtin_amdgcn_s_wait_tensorcnt(1); break;
        case 2: __builtin_amdgcn_s_wait_tensorcnt(2); break;
        case 3: __builtin_amdgcn_s_wait_tensorcnt(3); break;
        case 4: __builtin_amdgcn_s_wait_tensorcnt(4); break;
        default: __builtin_amdgcn_s_wait_tensorcnt(0); break;
    }
}
#endif // HAVE_TDM

__device__ __forceinline__ float clip01(float v) {
    return fminf(fmaxf(v, 0.0f), 1.0f);
}

__device__ __forceinline__ void biquad_coeffs(float fc, float qm,
        float& b0, float& b1, float& b2, float& a1, float& a2) {
    float w0 = MINW * __expf(fc * LN_WR);
    float q  = MINQ * __expf(qm * LN_QR);
    float sw = __sinf(w0);
    float cw = __cosf(w0);
    float al = sw / (2.0f * q);
    float ra = 1.0f / (1.0f + al);
    b1 = (1.0f - cw) * ra;
    b0 = 0.5f * b1;
    b2 = b0;
    a1 = -2.0f * cw * ra;
    a2 = (1.0f - al) * ra;
}

// Analytic VCO * OSC_GAIN * ADSR at sample n (n<0 -> 0).
__device__ __forceinline__ float sample_x(int n, float inc, float partials, float shape,
        float attack, float decay, float sustain, float release, float noteon,
        float alpha, float& env_out) {
    if (n < 0) { env_out = 0.0f; return 0.0f; }
    float u = inc * (float)(n + 1);
    u -= floorf(u);
    float ph = TWOPI * u;
    float sp = __sinf(ph);
    float cp = __cosf(ph);
    float sq = tanhf(1.57079632679f * partials * sp);
    float vco = (1.0f - 0.5f * shape) * sq * (1.0f + shape * cp);
    float t = (float)n * (1.0f / SRf);
    float att = __powf(clip01(t / attack), alpha);
    float dec = 1.0f - (1.0f - sustain) * __powf(clip01((t - attack) / decay), alpha);
    float rel = 1.0f - __powf(clip01((t - noteon) / release), alpha);
    float env = att * dec * rel;
    env_out = env;
    return vco * 0.5f * env;
}

// ---------------------------------------------------------------------------
// K1: thread = one chunk of 128 samples. fc/qm staged in via TDM, dry/env
// staged out via TDM (double-buffered, one issuing wave, s_wait_tensorcnt
// thresholds exploit TDM's in-order completion). Chunk affine transform
// (2x2 M, vec V) composed serially and written to ws.
// ---------------------------------------------------------------------------
__global__ void __launch_bounds__(RPB)
k_synth_compose(const float* __restrict__ f0_, const float* __restrict__ shape_,
                const float* __restrict__ noteon_, const float* __restrict__ fc_,
                const float* __restrict__ qm_, const float* __restrict__ attack_,
                const float* __restrict__ decay_, const float* __restrict__ sustain_,
                const float* __restrict__ release_, const float* __restrict__ alpha_,
                float* __restrict__ dry_out, float* __restrict__ env_out,
                float* __restrict__ Tws)
{
    __shared__ float sFC[2][RPB * PITCH];
    __shared__ float sQM[2][RPB * PITCH];
    __shared__ float sDR[2][RPB * PITCH];
    __shared__ float sEN[2][RPB * PITCH];

    int tid = threadIdx.x;
    int g = blockIdx.x * RPB + tid;
    int b = g >> 10;                       // NCH == 1024
    int c = g & (NCH - 1);
    int c0 = (blockIdx.x * RPB) & (NCH - 1);
    size_t rowoff = (size_t)b * NSAMP + (size_t)c0 * CHUNK;

    float f0 = f0_[b], shape = shape_[b], noteon = noteon_[b];
    float attack = attack_[b], decay = decay_[b], sustain = sustain_[b];
    float release = release_[b], alpha = alpha_[b];
    float inc = f0 * (1.0f / SRf);
    float partials = 12000.0f / (f0 * log10f(f0));

    int n0 = c * CHUNK;
    float envd;
    float x2 = sample_x(n0 - 2, inc, partials, shape, attack, decay, sustain, release, noteon, alpha, envd);
    float x1 = sample_x(n0 - 1, inc, partials, shape, attack, decay, sustain, release, noteon, alpha, envd);

    float M00 = 1.0f, M01 = 0.0f, M10 = 0.0f, M11 = 1.0f, V0 = 0.0f, V1 = 0.0f;

#if HAVE_TDM
    bool issuer = (tid < 32);              // wave 0 issues all DMA
    if (issuer) {
        tdm_issue(false, ldsoff(&sFC[0][0]), fc_ + rowoff);
        tdm_issue(false, ldsoff(&sQM[0][0]), qm_ + rowoff);
    }
#endif
    for (int t = 0; t < NTILES; ++t) {
        int buf = t & 1;
#if HAVE_TDM
        if (issuer) {
            if (t + 1 < NTILES) {
                tdm_issue(false, ldsoff(&sFC[buf ^ 1][0]), fc_ + rowoff + (t + 1) * TILE);
                tdm_issue(false, ldsoff(&sQM[buf ^ 1][0]), qm_ + rowoff + (t + 1) * TILE);
            }
            // in-order: everything up through loads(t) done when <= n remain
            wait_tensor(((t >= 1) ? 2 : 0) + ((t + 1 < NTILES) ? 2 : 0));
        }
        __syncthreads();
#else
        for (int k = tid; k < RPB * TILE; k += RPB) {
            int r = k >> 5, i = k & (TILE - 1);
            size_t go = rowoff + (size_t)r * CHUNK + t * TILE + i;
            sFC[buf][r * PITCH + i] = fc_[go];
            sQM[buf][r * PITCH + i] = qm_[go];
        }
        __syncthreads();
#endif
        for (int i = 0; i < TILE; ++i) {
            int n = n0 + t * TILE + i;
            float env;
            float x = sample_x(n, inc, partials, shape, attack, decay, sustain, release, noteon, alpha, env);
            sDR[buf][tid * PITCH + i] = x;
            sEN[buf][tid * PITCH + i] = env;
            float b0, b1, b2, a1, a2;
            biquad_coeffs(sFC[buf][tid * PITCH + i], sQM[buf][tid * PITCH + i],
                          b0, b1, b2, a1, a2);
            float f = b0 * x + b1 * x1 + b2 * x2;
            float nM00 = -a1 * M00 - a2 * M10;
            float nM01 = -a1 * M01 - a2 * M11;
            float nV0  = -a1 * V0  - a2 * V1 + f;
            M10 = M00; M11 = M01; V1 = V0;
            M00 = nM00; M01 = nM01; V0 = nV0;
            x2 = x1; x1 = x;
        }
        __syncthreads();
#if HAVE_TDM
        if (issuer) {
            tdm_issue(true, ldsoff(&sDR[buf][0]), dry_out + rowoff + t * TILE);
            tdm_issue(true, ldsoff(&sEN[buf][0]), env_out + rowoff + t * TILE);
        }
#else
        for (int k = tid; k < RPB * TILE; k += RPB) {
            int r = k >> 5, i = k & (TILE - 1);
            size_t go = rowoff + (size_t)r * CHUNK + t * TILE + i;
            dry_out[go] = sDR[buf][r * PITCH + i];
            env_out[go] = sEN[buf][r * PITCH + i];
        }
        __syncthreads();
#endif
    }
#if HAVE_TDM
    if (issuer) wait_tensor(0);
#endif
    Tws[0 * NCHUNKS + g] = M00;
    Tws[1 * NCHUNKS + g] = M01;
    Tws[2 * NCHUNKS + g] = V0;
    Tws[3 * NCHUNKS + g] = M10;
    Tws[4 * NCHUNKS + g] = M11;
    Tws[5 * NCHUNKS + g] = V1;
}

// ---------------------------------------------------------------------------
// K2: per-row Kogge-Stone scan over 1024 chunk transforms. Compositions are
// 3x3 homogeneous matrix products; 5 independent blocks packed on the
// diagonal of one exact V_WMMA_F32_16X16X4_F32 (C16x16 = A16x4 @ B4x16).
// ---------------------------------------------------------------------------
__global__ void __launch_bounds__(256)
k_scan_wmma(const float* __restrict__ Tws, float* __restrict__ S0, float* __restrict__ S1)
{
    __shared__ float Tb[2][6][NCH];        // 48 KB ping-pong
    int b = blockIdx.x;
    int tid = threadIdx.x;

    for (int c = tid; c < NCH; c += 256)
        for (int k = 0; k < 6; ++k)
            Tb[0][k][c] = Tws[k * NCHUNKS + b * NCH + c];
    __syncthreads();

    int wv   = tid >> 5;
    int lane = tid & 31;
    int l16  = lane & 15;
    bool hi  = lane >= 16;
    int jb   = l16 / 3;            // 3x3 block slot 0..4 (l16==15 unused)
    int rm   = l16 - 3 * jb;       // row-in-block (A) / col-in-block (B, D)

    int cur = 0;
    for (int d = 1; d < NCH; d <<= 1) {
        int nxt = cur ^ 1;
        for (int cp = tid; cp < d; cp += 256)      // pass-through elements
            for (int k = 0; k < 6; ++k)
                Tb[nxt][k][cp] = Tb[cur][k][cp];
        int nbatch = (NCH - d + 4) / 5;
        for (int bt = wv; bt < nbatch; bt += 8) {  // wave-uniform bound
            int idx  = bt * 5 + jb;
            int cpos = d + idx;
            bool valid = (jb < 5) && (cpos < NCH);
            float a0 = 0.0f, a1v = 0.0f, b0v = 0.0f, b1v = 0.0f;
            if (valid) {
                int cl = cpos;         // left  = current T[cpos]
                int cr = cpos - d;     // right = current T[cpos-d]
                if (!hi) {             // A: K=0,1
                    a0  = (rm < 2) ? Tb[cur][rm * 3 + 0][cl] : 0.0f;
                    a1v = (rm < 2) ? Tb[cur][rm * 3 + 1][cl] : 0.0f;
                } else {               // A: K=2,3
                    a0  = (rm < 2) ? Tb[cur][rm * 3 + 2][cl] : 1.0f;
                    a1v = 0.0f;
                }
                if (!hi) {             // B: rows 0 (v0) and 1 (v1)
                    b0v = Tb[cur][0 * 3 + rm][cr];
                    b1v = Tb[cur][1 * 3 + rm][cr];
                } else {               // B: rows 2 ([0 0 1]) and 3 (pad)
                    b0v = (rm == 2) ? 1.0f : 0.0f;
                    b1v = 0.0f;
                }
            }
            v2f Af = {a0, a1v};
            v2f Bf = {b0v, b1v};
            v8f Cz = {0.f, 0.f, 0.f, 0.f, 0.f, 0.f, 0.f, 0.f};
            v8f D = __builtin_amdgcn_wmma_f32_16x16x4_f32(
                        false, Af, false, Bf, (short)0, Cz, false, false);
            #pragma unroll
            for (int r = 0; r < 8; ++r) {
                int R  = r + (hi ? 8 : 0);
                int rr = R - 3 * jb;
                if (valid && rr >= 0 && rr < 2)
                    Tb[nxt][rr * 3 + rm][cpos] = D[r];
            }
        }
        __syncthreads();
        cur = nxt;
    }
    for (int c = tid; c < NCH; c += 256) {
        float s0 = 0.0f, s1 = 0.0f;
        if (c > 0) { s0 = Tb[cur][2][c - 1]; s1 = Tb[cur][5][c - 1]; }
        S0[b * NCH + c] = s0;
        S1[b * NCH + c] = s1;
    }
}

// ---------------------------------------------------------------------------
// K3: replay each chunk from its scanned start state. fc/qm/dry staged in,
// wet staged out via TDM. wet = tanh(y * gain).
// ---------------------------------------------------------------------------
__global__ void __launch_bounds__(RPB)
k_replay(const float* __restrict__ fc_, const float* __restrict__ qm_,
         const float* __restrict__ gain_, const float* __restrict__ dry_in,
         const float* __restrict__ S0, const float* __restrict__ S1,
         float* __restrict__ wet_out)
{
    __shared__ float sFC[2][RPB * PITCH];
    __shared__ float sQM[2][RPB * PITCH];
    __shared__ float sDY[2][RPB * PITCH];
    __shared__ float sWT[2][RPB * PITCH];

    int tid = threadIdx.x;
    int g = blockIdx.x * RPB + tid;
    int b = g >> 10;
    int c = g & (NCH - 1);
    int c0 = (blockIdx.x * RPB) & (NCH - 1);
    size_t rowoff = (size_t)b * NSAMP + (size_t)c0 * CHUNK;
    int n0 = c * CHUNK;

    float gain = gain_[b];
    float y1 = S0[g], y2 = S1[g];
    const float* dp = dry_in + (size_t)b * NSAMP + n0;
    float x1 = (n0 >= 1) ? dp[-1] : 0.0f;
    float x2 = (n0 >= 2) ? dp[-2] : 0.0f;

#if HAVE_TDM
    bool issuer = (tid < 32);
    if (issuer) {
        tdm_issue(false, ldsoff(&sFC[0][0]), fc_ + rowoff);
        tdm_issue(false, ldsoff(&sQM[0][0]), qm_ + rowoff);
        tdm_issue(false, ldsoff(&sDY[0][0]), dry_in + rowoff);
    }
#endif
    for (int t = 0; t < NTILES; ++t) {
        int buf = t & 1;
#if HAVE_TDM
        if (issuer) {
            if (t + 1 < NTILES) {
                tdm_issue(false, ldsoff(&sFC[buf ^ 1][0]), fc_ + rowoff + (t + 1) * TILE);
                tdm_issue(false, ldsoff(&sQM[buf ^ 1][0]), qm_ + rowoff + (t + 1) * TILE);
                tdm_issue(false, ldsoff(&sDY[buf ^ 1][0]), dry_in + rowoff + (t + 1) * TILE);
            }
            wait_tensor(((t >= 1) ? 1 : 0) + ((t + 1 < NTILES) ? 3 : 0));
        }
        __syncthreads();
#else
        for (int k = tid; k < RPB * TILE; k += RPB) {
            int r = k >> 5, i = k & (TILE - 1);
            size_t go = rowoff + (size_t)r * CHUNK + t * TILE + i;
            sFC[buf][r * PITCH + i] = fc_[go];
            sQM[buf][r * PITCH + i] = qm_[go];
            sDY[buf][r * PITCH + i] = dry_in[go];
        }
        __syncthreads();
#endif
        for (int i = 0; i < TILE; ++i) {
            float x = sDY[buf][tid * PITCH + i];
            float b0, b1, b2, a1, a2;
            biquad_coeffs(sFC[buf][tid * PITCH + i], sQM[buf][tid * PITCH + i],
                          b0, b1, b2, a1, a2);
            float y = b0 * x + b1 * x1 + b2 * x2 - a1 * y1 - a2 * y2;
            sWT[buf][tid * PITCH + i] = tanhf(y * gain);
            y2 = y1; y1 = y;
            x2 = x1; x1 = x;
        }
        __syncthreads();
#if HAVE_TDM
        if (issuer)
            tdm_issue(true, ldsoff(&sWT[buf][0]), wet_out + rowoff + t * TILE);
#else
        for (int k = tid; k < RPB * TILE; k += RPB) {
            int r = k >> 5, i = k & (TILE - 1);
            wet_out[rowoff + (size_t)r * CHUNK + t * TILE + i] = sWT[buf][r * PITCH + i];
        }
        __syncthreads();
#endif
    }
#if HAVE_TDM
    if (issuer) wait_tensor(0);
#endif
}

extern "C" void kernel_launch(void* const* d_in, const int* in_sizes, int n_in,
                              void* d_out, int out_size, void* d_ws, size_t ws_size,
                              hipStream_t stream) {
    (void)in_sizes; (void)n_in; (void)out_size; (void)ws_size;
    const float* f0      = (const float*)d_in[0];
    const float* shape   = (const float*)d_in[1];
    const float* noteon  = (const float*)d_in[2];
    const float* fc      = (const float*)d_in[3];
    const float* qm      = (const float*)d_in[4];
    const float* gain    = (const float*)d_in[5];
    const float* attack  = (const float*)d_in[6];
    const float* decay   = (const float*)d_in[7];
    const float* sustain = (const float*)d_in[8];
    const float* release = (const float*)d_in[9];
    const float* alpha   = (const float*)d_in[10];

    float* dry = (float*)d_out;
    float* wet = dry + (size_t)BATCH * NSAMP;
    float* env = wet + (size_t)BATCH * NSAMP;

    float* Tws = (float*)d_ws;                 // 6 * 32768 floats
    float* S0  = Tws + 6 * NCHUNKS;            // 32768 floats
    float* S1  = S0 + NCHUNKS;                 // 32768 floats

    k_synth_compose<<<NCHUNKS / RPB, RPB, 0, stream>>>(
        f0, shape, noteon, fc, qm, attack, decay, sustain, release, alpha,
        dry, env, Tws);
    k_scan_wmma<<<BATCH, 256, 0, stream>>>(Tws, S0, S1);
    k_replay<<<NCHUNKS / RPB, RPB, 0, stream>>>(fc, qm, gain, dry, S0, S1, wet);
}